// GNN_25847113187709
// MI455X (gfx1250) — compile-verified
//
#include <hip/hip_runtime.h>
#include <hip/hip_bf16.h>

typedef float v2f __attribute__((ext_vector_type(2)));
typedef float v8f __attribute__((ext_vector_type(8)));

#define N_NODES 50000
#define KDIM    128
#define LDA     132   // padded LDS row stride (floats): 132 % 64 = 4 banks/row -> conflict-free

// ---------------------------------------------------------------------------
// deg[dst] += 1 per edge (hardware fp32 atomic)
// ---------------------------------------------------------------------------
__global__ __launch_bounds__(256) void deg_kernel(const int* __restrict__ dst,
                                                  float* __restrict__ deg, int nE) {
    int e = blockIdx.x * blockDim.x + threadIdx.x;
    if (e < nE) unsafeAtomicAdd(&deg[dst[e]], 1.0f);
}

// ---------------------------------------------------------------------------
// msg[dst] += feat[src] : 32 lanes per edge, float4 per lane, HW fp32 atomics
// ---------------------------------------------------------------------------
__global__ __launch_bounds__(256) void scatter_kernel(const float* __restrict__ feat,
                                                      const int* __restrict__ src,
                                                      const int* __restrict__ dst,
                                                      float* __restrict__ msg, int nE) {
    int t = blockIdx.x * blockDim.x + threadIdx.x;
    int e = t >> 5;
    if (e >= nE) return;
    int c = (t & 31) << 2;                       // feature offset 0..124
    int s = src[e], d = dst[e];
    const float4 v = *(const float4*)(feat + (size_t)s * KDIM + c);
    float* p = msg + (size_t)d * KDIM + c;
    unsafeAtomicAdd(p + 0, v.x);
    unsafeAtomicAdd(p + 1, v.y);
    unsafeAtomicAdd(p + 2, v.z);
    unsafeAtomicAdd(p + 3, v.w);
}

// ---------------------------------------------------------------------------
// Fused: A = (msg + feat) / (deg + 1)   (16 x 128 tile staged in LDS)
//        out[16 x n_out] = A @ W + bias  (optional leaky-relu)
// One workgroup per 16-row tile; wave w computes output cols [16w, 16w+16).
// Uses V_WMMA_F32_16X16X4_F32 (full fp32 precision on the matrix pipe).
//
// fp32 WMMA VGPR layouts (ISA 7.12.2):
//   A 16x4 : lane<16 -> row=lane,    v0=K+0, v1=K+1
//            lane>=16-> row=lane-16, v0=K+2, v1=K+3
//   B 4x16 : lane<16 -> col=lane,    v0=K+0, v1=K+1
//            lane>=16-> col=lane-16, v0=K+2, v1=K+3
//   C/D    : lane<16 -> col=lane,    acc[i] = row i
//            lane>=16-> col=lane-16, acc[i] = row 8+i
// ---------------------------------------------------------------------------
__global__ __launch_bounds__(256) void sage_gemm_wmma(const float* __restrict__ msg,
                                                      const float* __restrict__ feat,
                                                      const float* __restrict__ deg,
                                                      const float* __restrict__ W,    // [128 x n_out]
                                                      const float* __restrict__ bias, // [n_out]
                                                      float* __restrict__ out,        // [N x n_out]
                                                      int n_out, int apply_lrelu) {
    __shared__ float Atile[16 * LDA];

    const int tid  = threadIdx.x;
    const int nt   = blockDim.x;
    const int wave = tid >> 5;
    const int lane = tid & 31;
    const int m0   = blockIdx.x * 16;

    // --- cooperative load + normalize of the 16x128 A tile (float4 granular) ---
    for (int i = tid; i < 16 * (KDIM / 4); i += nt) {
        int r  = i >> 5;              // row in tile (KDIM/4 == 32)
        int c4 = (i & 31) << 2;       // column (multiple of 4)
        int row = m0 + r;
        float inv = 1.0f / (deg[row] + 1.0f);
        const float4 mv = *(const float4*)(msg  + (size_t)row * KDIM + c4);
        const float4 xv = *(const float4*)(feat + (size_t)row * KDIM + c4);
        float* a = &Atile[r * LDA + c4];
        a[0] = (mv.x + xv.x) * inv;
        a[1] = (mv.y + xv.y) * inv;
        a[2] = (mv.z + xv.z) * inv;
        a[3] = (mv.w + xv.w) * inv;
    }
    __syncthreads();

    // --- per-wave 16x16 output tile via fp32 WMMA ---
    const int n0 = wave * 16;
    const int lr = lane & 15;
    const int hi = lane >> 4;         // 0 or 1 -> K sub-offset 2*hi

    v8f acc = {0.f, 0.f, 0.f, 0.f, 0.f, 0.f, 0.f, 0.f};

#pragma unroll 8
    for (int k = 0; k < KDIM; k += 4) {
        v2f a, b;
        const float* ap = &Atile[lr * LDA + k + 2 * hi];
        a.x = ap[0];
        a.y = ap[1];
        const float* bp = &W[(size_t)(k + 2 * hi) * n_out + n0 + lr];
        b.x = bp[0];
        b.y = bp[n_out];
        acc = __builtin_amdgcn_wmma_f32_16x16x4_f32(
            /*neg_a=*/false, a, /*neg_b=*/false, b,
            /*c_mod=*/(short)0, acc, /*reuse_a=*/false, /*reuse_b=*/false);
    }

    // --- epilogue: bias (+ leaky relu), store ---
    const float bv   = bias[n0 + lr];
    const int  rbase = m0 + 8 * hi;
    const int  col   = n0 + lr;
#pragma unroll
    for (int i = 0; i < 8; ++i) {
        float v = acc[i] + bv;
        if (apply_lrelu) v = (v >= 0.f) ? v : 0.01f * v;
        out[(size_t)(rbase + i) * n_out + col] = v;
    }
}

// ---------------------------------------------------------------------------
extern "C" void kernel_launch(void* const* d_in, const int* in_sizes, int n_in,
                              void* d_out, int out_size, void* d_ws, size_t ws_size,
                              hipStream_t stream) {
    const float* x   = (const float*)d_in[0];          // [50000 x 128]
    const int*   edg = (const int*)  d_in[1];          // [2 x E]
    const float* W1  = (const float*)d_in[2];          // [128 x 128]
    const float* b1  = (const float*)d_in[3];          // [128]
    const float* W2  = (const float*)d_in[4];          // [128 x 32]
    const float* b2  = (const float*)d_in[5];          // [32]
    float*       outp = (float*)d_out;                 // [50000 x 32]

    const int E = in_sizes[1] / 2;
    const int* src = edg;
    const int* dst = edg + E;

    // workspace: msg [N*128] | deg [N] | h1 [N*128]
    float* msg = (float*)d_ws;
    float* deg = msg + (size_t)N_NODES * KDIM;
    float* h1  = deg + N_NODES;

    const size_t msgBytes = (size_t)N_NODES * KDIM * sizeof(float);

    // ---- layer 1 ----
    hipMemsetAsync(msg, 0, msgBytes, stream);
    hipMemsetAsync(deg, 0, (size_t)N_NODES * sizeof(float), stream);

    deg_kernel<<<(E + 255) / 256, 256, 0, stream>>>(dst, deg, E);

    {
        long long threads = (long long)E * 32;
        int blocks = (int)((threads + 255) / 256);
        scatter_kernel<<<blocks, 256, 0, stream>>>(x, src, dst, msg, E);
    }

    // 50000 / 16 = 3125 row tiles exactly; 8 waves -> 128 output cols
    sage_gemm_wmma<<<N_NODES / 16, 256, 0, stream>>>(msg, x, deg, W1, b1, h1,
                                                     /*n_out=*/128, /*lrelu=*/1);

    // ---- layer 2 ----
    hipMemsetAsync(msg, 0, msgBytes, stream);

    {
        long long threads = (long long)E * 32;
        int blocks = (int)((threads + 255) / 256);
        scatter_kernel<<<blocks, 256, 0, stream>>>(h1, src, dst, msg, E);
    }

    // 2 waves -> 32 output cols
    sage_gemm_wmma<<<N_NODES / 16, 64, 0, stream>>>(msg, h1, deg, W2, b2, outp,
                                                    /*n_out=*/32, /*lrelu=*/0);
}